// HSSurv_12429635355022
// MI455X (gfx1250) — compile-verified
//
#include <hip/hip_runtime.h>
#include <math.h>

// ---------------- problem constants (match reference) ----------------
#define BSZ   8
#define NTOK  2048
#define CDIM  512
#define KEXP  8
#define EPSV  1e-6f
#define GENO_RATIO 0.1f

typedef __attribute__((ext_vector_type(2))) float v2f;
typedef __attribute__((ext_vector_type(8))) float v8f;

// workspace layout (units: 4-byte words)
#define WS_COUNTS 0                       // 64 int   (zeroed)
#define WS_MASS   64                      // 64 float (zeroed)
#define WS_HSUM   128                     // 32768 float (zeroed)
#define WS_GENO   (128 + 32768)           // 64 float
#define WS_IDX    (WS_GENO + 64)          // B*K*N = 131072 int
#define WS_WGT    (WS_IDX + BSZ*KEXP*NTOK)// 131072 float

#define SPLITS 8          // tile-splits per (b,k) group in expert1
#define APITCH 516        // LDS row pitch in floats (2064 B; bank-conflict pad)

// ---------------------------------------------------------------
// geno logits: genoLog[b,k] = dot(geno_vec[b], geno_w[:,k]) + geno_b[k]
// ---------------------------------------------------------------
__global__ __launch_bounds__(256) void geno_kernel(
    const float* __restrict__ geno_vec, const float* __restrict__ geno_w,
    const float* __restrict__ geno_b, float* __restrict__ genoLog) {
  int b = blockIdx.x;
  int lane = threadIdx.x & 31;
  int k = threadIdx.x >> 5;            // 8 waves -> 8 experts
  float acc = 0.f;
  for (int c = lane; c < CDIM; c += 32)
    acc += geno_vec[b * CDIM + c] * geno_w[c * KEXP + k];
#pragma unroll
  for (int off = 16; off > 0; off >>= 1) acc += __shfl_xor(acc, off, 32);
  if (lane == 0) genoLog[b * KEXP + k] = acc + geno_b[k];
}

// ---------------------------------------------------------------
// gating: one wave per token. logits -> top2 -> softmax -> scatter into
// per-(b,k) gather lists + mass sums. counts double as usage hit counts.
// ---------------------------------------------------------------
__global__ __launch_bounds__(256) void gate_kernel(
    const float* __restrict__ tokens, const float* __restrict__ gate_w,
    const float* __restrict__ gate_b, const float* __restrict__ genoLog,
    int* __restrict__ counts, int* __restrict__ idxL,
    float* __restrict__ wgtL, float* __restrict__ mass) {
  int tid = threadIdx.x;
  int lane = tid & 31;
  int wv = tid >> 5;
  int tok = blockIdx.x * 8 + wv;             // 8 tokens per block
  int b = tok >> 11;                         // NTOK = 2048
  int n = tok & (NTOK - 1);

  const float* xp = tokens + (size_t)tok * CDIM;
  float acc[KEXP];
#pragma unroll
  for (int k = 0; k < KEXP; ++k) acc[k] = 0.f;

  for (int c = lane; c < CDIM; c += 32) {
    float x = xp[c];
    const float4* gw = (const float4*)(gate_w + c * KEXP);
    float4 g0 = gw[0], g1 = gw[1];
    acc[0] += x * g0.x; acc[1] += x * g0.y; acc[2] += x * g0.z; acc[3] += x * g0.w;
    acc[4] += x * g1.x; acc[5] += x * g1.y; acc[6] += x * g1.z; acc[7] += x * g1.w;
  }
#pragma unroll
  for (int k = 0; k < KEXP; ++k) {
#pragma unroll
    for (int off = 16; off > 0; off >>= 1) acc[k] += __shfl_xor(acc[k], off, 32);
  }

  if (lane == 0) {
    float lg[KEXP];
#pragma unroll
    for (int k = 0; k < KEXP; ++k)
      lg[k] = acc[k] + gate_b[k] + GENO_RATIO * genoLog[b * KEXP + k];
    // GATE_TEMP == 1.0 -> no-op divide

    // top-2 (stable: first index wins ties, matches jax top_k)
    int i0 = 0; float v0 = lg[0];
#pragma unroll
    for (int k = 1; k < KEXP; ++k)
      if (lg[k] > v0) { v0 = lg[k]; i0 = k; }
    int i1 = -1; float v1 = 0.f;
#pragma unroll
    for (int k = 0; k < KEXP; ++k) {
      if (k == i0) continue;
      if (i1 < 0 || lg[k] > v1) { v1 = lg[k]; i1 = k; }
    }
    // softmax over (v0, v1), v0 >= v1
    float e = expf(v1 - v0);
    float w0 = 1.f / (1.f + e);
    float w1 = e / (1.f + e);
    w0 = fmaxf(w0, EPSV); w1 = fmaxf(w1, EPSV);
    float s = w0 + w1; w0 /= s; w1 /= s;

    int g0i = b * KEXP + i0;
    int p0 = atomicAdd(&counts[g0i], 1);
    idxL[g0i * NTOK + p0] = n;
    wgtL[g0i * NTOK + p0] = w0;
    atomicAdd(&mass[g0i], w0);

    int g1i = b * KEXP + i1;
    int p1 = atomicAdd(&counts[g1i], 1);
    idxL[g1i * NTOK + p1] = n;
    wgtL[g1i * NTOK + p1] = w1;
    atomicAdd(&mass[g1i], w1);
  }
}

// ---------------------------------------------------------------
// expert layer 1 + weighted combine:
//   hsum[b,k,c] = sum_{selected n} W[b,n,k] * relu(tokens[b,n]@w1[k] + b1[k])[c]
// grid = (64 groups, SPLITS). Each block handles tiles of 16 gathered tokens.
// A tile staged via async global->LDS DMA (GLOBAL_LOAD_ASYNC_TO_LDS_B128,
// ASYNCcnt); 8 waves x 64 output columns each; WMMA F32 16x16x4.
// Pad rows (>= count) alias a valid token row; their gate weight is 0, so
// they contribute exactly 0 after the weighted reduce.
// ---------------------------------------------------------------
__global__ __launch_bounds__(256) void expert1_kernel(
    const float* __restrict__ tokens, const float* __restrict__ w1,
    const float* __restrict__ b1, const int* __restrict__ counts,
    const int* __restrict__ idxL, const float* __restrict__ wgtL,
    float* __restrict__ hsum) {
  __shared__ float At[16 * APITCH];   // 16 gathered token rows x 512 (padded)
  __shared__ float colAcc[CDIM];      // per-block column accumulator
  __shared__ float wt[16];            // per-row gate weights

  int g = blockIdx.x;                 // b*K + k
  int split = blockIdx.y;
  int b = g >> 3;
  int k = g & 7;

  int tid = threadIdx.x;
  int lane = tid & 31;
  int wv = tid >> 5;                  // 8 waves
  int half = lane >> 4;               // 0: lanes 0-15, 1: lanes 16-31
  int m = lane & 15;
  int col0base = wv * 64;             // this wave owns columns [col0base, col0base+64)

  for (int i = tid; i < CDIM; i += 256) colAcc[i] = 0.f;

  int cnt = counts[g];
  int ntiles = (cnt + 15) >> 4;
  const float* w1k = w1 + (size_t)k * CDIM * CDIM;
  const float* b1k = b1 + k * CDIM;

  // LDS byte offset of At within this workgroup's allocation
  unsigned atBase = (unsigned)(uintptr_t)(&At[0]);

  for (int t = split; t < ntiles; t += SPLITS) {
    __syncthreads();  // previous-iteration readers done with At/colAcc init
    int base = t * 16;
    if (tid < 16)
      wt[tid] = (base + tid < cnt) ? wgtL[g * NTOK + base + tid] : 0.f;

    // ---- async DMA: 2 rows per wave, 4 x B128 per row (2 KB/row) ----
#pragma unroll
    for (int rr = 0; rr < 2; ++rr) {
      int row = wv * 2 + rr;
      int ridx = base + row;
      int n = idxL[g * NTOK + (ridx < cnt ? ridx : cnt - 1)];
      unsigned long long ga =
          (unsigned long long)(uintptr_t)(tokens + (size_t)(b * NTOK + n) * CDIM) +
          (unsigned)(lane * 16);
      unsigned ldso = atBase + (unsigned)(row * APITCH * 4 + lane * 16);
      asm volatile(
          "global_load_async_to_lds_b128 %0, %1, off\n\t"
          "global_load_async_to_lds_b128 %0, %1, off offset:512\n\t"
          "global_load_async_to_lds_b128 %0, %1, off offset:1024\n\t"
          "global_load_async_to_lds_b128 %0, %1, off offset:1536"
          :: "v"(ldso), "v"(ga) : "memory");
    }
    asm volatile("s_wait_asynccnt 0x0" ::: "memory");
    __syncthreads();

    v8f acc[4] = {};  // 4 column sub-tiles of 16
    const float* arow = &At[m * APITCH];
    for (int kc = 0; kc < CDIM; kc += 4) {
      int r0 = kc + 2 * half;        // A frag: lanes 0-15 K=kc+0/1, lanes 16-31 K=kc+2/3
      v2f a;
      a.x = arow[r0];
      a.y = arow[r0 + 1];
      const float* bp = w1k + (size_t)r0 * CDIM + col0base + m;
#pragma unroll
      for (int ct = 0; ct < 4; ++ct) {
        v2f bb;                      // B frag mirrored: lane = N, K=r0/r0+1
        bb.x = bp[ct * 16];
        bb.y = bp[ct * 16 + CDIM];
        acc[ct] = __builtin_amdgcn_wmma_f32_16x16x4_f32(
            false, a, false, bb, (short)0, acc[ct], false, false);
      }
    }

    // epilogue: bias + relu + gate-weight, reduce over 16 rows in-fragment
#pragma unroll
    for (int ct = 0; ct < 4; ++ct) {
      int col = col0base + ct * 16 + m;
      float bias = b1k[col];
      float part = 0.f;
#pragma unroll
      for (int r = 0; r < 8; ++r) {   // C layout: VGPR r holds M = half*8 + r, N = m
        float v = acc[ct][r] + bias;
        v = v > 0.f ? v : 0.f;
        part += wt[half * 8 + r] * v;
      }
      part += __shfl_xor(part, 16, 32);   // combine M=0..7 with M=8..15
      if (half == 0) colAcc[col] += part; // wave-private column range
    }
  }

  __syncthreads();
  for (int i = tid; i < CDIM; i += 256)
    atomicAdd(&hsum[g * CDIM + i], colAcc[i]);
}

// ---------------------------------------------------------------
// expert layer 2 + normalize:
//   centers[b,k,e] = (hsum[b,k]@w2[k] + b2[k]*mass) / max(mass, eps)
// ---------------------------------------------------------------
__global__ __launch_bounds__(256) void expert2_kernel(
    const float* __restrict__ w2, const float* __restrict__ b2,
    const float* __restrict__ hsum, const float* __restrict__ mass,
    float* __restrict__ out) {
  __shared__ float hl[CDIM];
  int g = blockIdx.x;
  int k = g & 7;
  int tid = threadIdx.x;
  for (int i = tid; i < CDIM; i += 256) hl[i] = hsum[g * CDIM + i];
  __syncthreads();
  float ms = mass[g];
  float inv = 1.f / fmaxf(ms, EPSV);
  const float* w2k = w2 + (size_t)k * CDIM * CDIM;
  for (int e = tid; e < CDIM; e += 256) {
    float acc = 0.f;
#pragma unroll 4
    for (int c = 0; c < CDIM; ++c) acc = fmaf(hl[c], w2k[c * CDIM + e], acc);
    out[g * CDIM + e] = (acc + b2[k * CDIM + e] * ms) * inv;
  }
}

// ---------------------------------------------------------------
// load-balance loss: cv^2 of expert usage (population std)
// ---------------------------------------------------------------
__global__ void lb_kernel(const int* __restrict__ counts, float* __restrict__ out) {
  if (threadIdx.x == 0 && blockIdx.x == 0) {
    float usage[KEXP];
    float mean = 0.f;
    for (int k = 0; k < KEXP; ++k) {
      int t = 0;
      for (int b = 0; b < BSZ; ++b) t += counts[b * KEXP + k];
      usage[k] = (float)t / (float)(BSZ * NTOK);
      mean += usage[k];
    }
    mean /= (float)KEXP;
    float var = 0.f;
    for (int k = 0; k < KEXP; ++k) {
      float d = usage[k] - mean;
      var += d * d;
    }
    var /= (float)KEXP;
    float denom = mean + EPSV;
    out[BSZ * KEXP * CDIM] = var / (denom * denom);
  }
}

// ---------------------------------------------------------------
extern "C" void kernel_launch(void* const* d_in, const int* in_sizes, int n_in,
                              void* d_out, int out_size, void* d_ws, size_t ws_size,
                              hipStream_t stream) {
  const float* tokens   = (const float*)d_in[0];
  const float* geno_vec = (const float*)d_in[1];
  const float* gate_w   = (const float*)d_in[2];
  const float* gate_b   = (const float*)d_in[3];
  const float* geno_w   = (const float*)d_in[4];
  const float* geno_b   = (const float*)d_in[5];
  const float* w1       = (const float*)d_in[6];
  const float* b1       = (const float*)d_in[7];
  const float* w2       = (const float*)d_in[8];
  const float* b2       = (const float*)d_in[9];
  float* out = (float*)d_out;

  float* wsf = (float*)d_ws;
  int*   counts  = (int*)(wsf + WS_COUNTS);
  float* mass    = wsf + WS_MASS;
  float* hsum    = wsf + WS_HSUM;
  float* genoLog = wsf + WS_GENO;
  int*   idxL    = (int*)(wsf + WS_IDX);
  float* wgtL    = wsf + WS_WGT;

  // zero counters, mass sums, hsum accumulators (graph-capture safe)
  hipMemsetAsync(d_ws, 0, (size_t)(64 + 64 + BSZ * KEXP * CDIM) * sizeof(float), stream);

  geno_kernel<<<BSZ, 256, 0, stream>>>(geno_vec, geno_w, geno_b, genoLog);
  gate_kernel<<<(BSZ * NTOK) / 8, 256, 0, stream>>>(
      tokens, gate_w, gate_b, genoLog, counts, idxL, wgtL, mass);
  dim3 g1(BSZ * KEXP, SPLITS);
  expert1_kernel<<<g1, 256, 0, stream>>>(tokens, w1, b1, counts, idxL, wgtL, hsum);
  expert2_kernel<<<BSZ * KEXP, 256, 0, stream>>>(w2, b2, hsum, mass, out);
  lb_kernel<<<1, 32, 0, stream>>>(counts, out);
}